// GAT_pyg_48017734369598
// MI455X (gfx1250) — compile-verified
//
#include <hip/hip_runtime.h>
#include <math.h>

#define N_NODES  100000
#define N_EDGES  2000000
#define E_TOTAL  (N_EDGES + N_NODES)   // self-loops appended implicitly
#define DIM      32
#define NEG_SLOPE 0.2f
#define GAT_EPS  1e-16f

typedef __attribute__((ext_vector_type(2))) float v2f;
typedef __attribute__((ext_vector_type(8))) float v8f;

// ---- monotone float<->uint mapping for atomicMax on floats ----
__device__ __forceinline__ unsigned f2ord(float f) {
    unsigned u = __float_as_uint(f);
    return (u & 0x80000000u) ? ~u : (u | 0x80000000u);
}
__device__ __forceinline__ float ord2f(unsigned u) {
    return __uint_as_float((u & 0x80000000u) ? (u & 0x7FFFFFFFu) : ~u);
}
#define ORD_NEG_INF 0x007FFFFFu   // f2ord(-inf)

// =====================================================================
// GEMM H = X @ W  (X:[N,32] row-major, W:[32,32] row-major) via
// V_WMMA_F32_16X16X4_F32. One wave computes a 16x32 output tile.
// A 16x4 fp32 layout: lane m=lane&15; VGPR0/1 = K (k0 + (lane>=16?2:0)) +0/+1
// B 4x16  fp32 layout: col = lane&15 (+16 for tile 1); VGPR0/1 = rows K as above
// C 16x16 fp32 layout: VGPR r -> row r + (lane>=16 ? 8 : 0), col = lane&15
// =====================================================================
__global__ __launch_bounds__(32)
void gemm16_wmma(const float* __restrict__ X, const float* __restrict__ W,
                 float* __restrict__ H) {
    const int lane = threadIdx.x;
    const int half = lane >> 4;       // 0 | 1
    const int l15  = lane & 15;
    const int row0 = blockIdx.x * 16;

    // Preload all of W as B fragments: 2 n-tiles x 8 k-steps
    v2f bfrag[2][8];
#pragma unroll
    for (int nt = 0; nt < 2; ++nt) {
#pragma unroll
        for (int ks = 0; ks < 8; ++ks) {
            const int k   = ks * 4 + half * 2;
            const int col = nt * 16 + l15;
            bfrag[nt][ks].x = W[(k + 0) * DIM + col];
            bfrag[nt][ks].y = W[(k + 1) * DIM + col];
        }
    }

    v8f c0 = {}; v8f c1 = {};
    const int arow = row0 + l15;
#pragma unroll
    for (int ks = 0; ks < 8; ++ks) {
        const int k = ks * 4 + half * 2;
        v2f a;
        a.x = X[arow * DIM + k];
        a.y = X[arow * DIM + k + 1];
        c0 = __builtin_amdgcn_wmma_f32_16x16x4_f32(false, a, false, bfrag[0][ks],
                                                   (short)0, c0, false, false);
        c1 = __builtin_amdgcn_wmma_f32_16x16x4_f32(false, a, false, bfrag[1][ks],
                                                   (short)0, c1, false, false);
    }

#pragma unroll
    for (int r = 0; r < 8; ++r) {
        const int row = row0 + r + half * 8;
        H[row * DIM + l15]      = c0[r];
        H[row * DIM + 16 + l15] = c1[r];
    }
}

// wa[0:32) = W @ a_src, wa[32:64) = W @ a_dst   (1 block, 32 threads)
__global__ __launch_bounds__(32)
void compute_wa(const float* __restrict__ W, const float* __restrict__ a_src,
                const float* __restrict__ a_dst, float* __restrict__ wa) {
    const int k = threadIdx.x;
    float s = 0.f, d = 0.f;
#pragma unroll
    for (int n = 0; n < DIM; ++n) {
        const float w = W[k * DIM + n];
        s += w * a_src[n];
        d += w * a_dst[n];
    }
    wa[k] = s;
    wa[32 + k] = d;
}

// alpha_s[i] = x[i,:].wa_src, alpha_d[i] = x[i,:].wa_dst; wave per node, lane=dim
__global__ __launch_bounds__(256)
void node_alpha(const float* __restrict__ X, const float* __restrict__ wa,
                float* __restrict__ as, float* __restrict__ ad) {
    const int lane = threadIdx.x & 31;
    const int node = (blockIdx.x * blockDim.x + threadIdx.x) >> 5;
    if (node >= N_NODES) return;
    const float xv = X[node * DIM + lane];
    float s = xv * wa[lane];
    float d = xv * wa[32 + lane];
#pragma unroll
    for (int m = 16; m; m >>= 1) {
        s += __shfl_xor(s, m, 32);
        d += __shfl_xor(d, m, 32);
    }
    if (lane == 0) { as[node] = s; ad[node] = d; }
}

__global__ __launch_bounds__(256)
void init_nodes(unsigned* __restrict__ m, float* __restrict__ denom) {
    const int i = blockIdx.x * blockDim.x + threadIdx.x;
    if (i < N_NODES) { m[i] = ORD_NEG_INF; denom[i] = 0.f; }
}

__global__ __launch_bounds__(256)
void zero_vec(float* __restrict__ p, int n) {
    const int i = blockIdx.x * blockDim.x + threadIdx.x;
    if (i < n) p[i] = 0.f;
}

__device__ __forceinline__ void edge_sd(const long long* __restrict__ ei, int e,
                                        int& s, int& d) {
    if (e < N_EDGES) { s = (int)ei[e]; d = (int)ei[N_EDGES + e]; }
    else             { s = d = e - N_EDGES; }            // self-loop
}

// pass 1: e = leaky_relu(as[s]+ad[d]); segment max over dst
__global__ __launch_bounds__(256)
void edge_max(const long long* __restrict__ ei, const float* __restrict__ as,
              const float* __restrict__ ad, float* __restrict__ ebuf,
              unsigned* __restrict__ m) {
    const int e = blockIdx.x * blockDim.x + threadIdx.x;
    if (e >= E_TOTAL) return;
    int s, d; edge_sd(ei, e, s, d);
    float v = as[s] + ad[d];
    v = v > 0.f ? v : NEG_SLOPE * v;
    ebuf[e] = v;
    atomicMax(m + d, f2ord(v));
}

// pass 2: exp(e - m[dst]); segment sum -> denom
__global__ __launch_bounds__(256)
void edge_exp(const long long* __restrict__ ei, const unsigned* __restrict__ m,
              float* __restrict__ ebuf, float* __restrict__ denom) {
    const int e = blockIdx.x * blockDim.x + threadIdx.x;
    if (e >= E_TOTAL) return;
    int s, d; edge_sd(ei, e, s, d); (void)s;
    const float ex = expf(ebuf[e] - ord2f(m[d]));
    ebuf[e] = ex;
    atomicAdd(denom + d, ex);
}

// pass 3: out[dst,:] += h[src,:] * alpha; one wave per edge, lane = dim
__global__ __launch_bounds__(256)
void edge_aggr(const long long* __restrict__ ei, const float* __restrict__ h,
               const float* __restrict__ ebuf, const float* __restrict__ denom,
               float* __restrict__ out) {
    const int lane = threadIdx.x & 31;
    const int e    = blockIdx.x * (blockDim.x >> 5) + (threadIdx.x >> 5);
    if (e >= E_TOTAL) return;
    int s, d; edge_sd(ei, e, s, d);
    const float alpha = ebuf[e] / (denom[d] + GAT_EPS);
    atomicAdd(out + d * DIM + lane, h[s * DIM + lane] * alpha);
}

// out = elu(out + bias)
__global__ __launch_bounds__(256)
void bias_elu(float* __restrict__ io, const float* __restrict__ bias) {
    const int i = blockIdx.x * blockDim.x + threadIdx.x;
    if (i >= N_NODES * DIM) return;
    const float v = io[i] + bias[i & (DIM - 1)];
    io[i] = v > 0.f ? v : (expf(v) - 1.f);
}

// =====================================================================
static void run_layer(const float* X, const long long* ei, const float* W,
                      const float* a_src, const float* a_dst, const float* bias,
                      float* out, float* ws, hipStream_t stream) {
    float*    H    = ws;                      // N*32
    float*    AS   = H  + N_NODES * DIM;      // N
    float*    AD   = AS + N_NODES;            // N
    unsigned* M    = (unsigned*)(AD + N_NODES);   // N (uint)
    float*    DEN  = (float*)(M + N_NODES);   // N
    float*    EBUF = DEN + N_NODES;           // E_TOTAL
    float*    WA   = EBUF + E_TOTAL;          // 64

    const int nodeBlk  = (N_NODES + 255) / 256;
    const int featBlk  = (N_NODES * DIM + 255) / 256;
    const int edgeBlk  = (E_TOTAL + 255) / 256;
    const int edgeWave = (E_TOTAL + 7) / 8;   // 8 waves of 32 per block

    compute_wa<<<1, 32, 0, stream>>>(W, a_src, a_dst, WA);
    gemm16_wmma<<<N_NODES / 16, 32, 0, stream>>>(X, W, H);
    node_alpha<<<(N_NODES * 32 + 255) / 256, 256, 0, stream>>>(X, WA, AS, AD);
    init_nodes<<<nodeBlk, 256, 0, stream>>>(M, DEN);
    zero_vec<<<featBlk, 256, 0, stream>>>(out, N_NODES * DIM);
    edge_max<<<edgeBlk, 256, 0, stream>>>(ei, AS, AD, EBUF, M);
    edge_exp<<<edgeBlk, 256, 0, stream>>>(ei, M, EBUF, DEN);
    edge_aggr<<<edgeWave, 256, 0, stream>>>(ei, H, EBUF, DEN, out);
    bias_elu<<<featBlk, 256, 0, stream>>>(out, bias);
}

extern "C" void kernel_launch(void* const* d_in, const int* in_sizes, int n_in,
                              void* d_out, int out_size, void* d_ws, size_t ws_size,
                              hipStream_t stream) {
    (void)in_sizes; (void)n_in; (void)out_size; (void)ws_size;
    const float*     x   = (const float*)d_in[0];
    const long long* ei  = (const long long*)d_in[1];   // int64 edge_index [2,E]
    const float*     W1  = (const float*)d_in[2];
    const float*     as1 = (const float*)d_in[3];
    const float*     ad1 = (const float*)d_in[4];
    const float*     b1  = (const float*)d_in[5];
    const float*     W2  = (const float*)d_in[6];
    const float*     as2 = (const float*)d_in[7];
    const float*     ad2 = (const float*)d_in[8];
    const float*     b2  = (const float*)d_in[9];

    float* xbar = (float*)d_out;              // output 0: elu(gat2(z))
    float* z    = xbar + N_NODES * DIM;       // output 1: elu(gat1(x))
    float* ws   = (float*)d_ws;

    run_layer(x, ei, W1, as1, ad1, b1, z,    ws, stream);
    run_layer(z, ei, W2, as2, ad2, b2, xbar, ws, stream);
}